// SqRL_44332652429603
// MI455X (gfx1250) — compile-verified
//
#include <hip/hip_runtime.h>

// Ring-unroll gather for SquareRotationalLayer on gfx1250 (MI455X).
// x (16,32,512,512) f32 -> out (16,32,256,2048) f32.
// Pure bandwidth kernel: ~1.5 GiB HBM traffic/call, ~65us floor at 23.3 TB/s.
//
// Branchless index math: the ring-perimeter walk for output row `dif` is
//   i = 255-dif, el = 2*dif+1, j = i+el = 511-i
//   c = i + clamp(t-b0,0,el) - clamp(t-b4,0,el)      (b0=i,    b4=i+1022)
//   r = i + clamp(t-b2,0,el) - clamp(t-b6,0,el)      (b2=i+511, b6=i+1533)
// which reproduces all 9 segments (corner repeats + 4 edges) exactly.
// clamp(x,0,el) lowers to a single v_med3_i32 -> no divergence at all.

constexpr int H       = 512;
constexpr int W       = 512;
constexpr int NROWS   = 256;   // H/2 output rows per image
constexpr int NCOLS   = 2048;  // 4*H output cols per row
constexpr int RINGLEN = 2044;  // 4*H - 4 (last 4 cols replicate cols 0..3)

// Native clang vector type: accepted by __builtin_nontemporal_store and lowers
// to a single b128 access.
typedef __attribute__((ext_vector_type(4))) float v4f;

__device__ __forceinline__ int iclamp0(int x, int hi) {
    return min(max(x, 0), hi);           // v_med3_i32
}

__device__ __forceinline__ int ring_src(int i, int el, int ceff) {
    const int t0 = ceff - i;             // t - b0
    const int m0 = iclamp0(t0,        el);
    const int m2 = iclamp0(t0 - 511,  el);
    const int m4 = iclamp0(t0 - 1022, el);
    const int m6 = iclamp0(t0 - 1533, el);
    const int c  = i + m0 - m4;
    const int r  = i + m2 - m6;
    return (r << 9) + c;                 // r*W + c, W = 512
}

__global__ __launch_bounds__(256) void SqRL_ring_gather(
    const float* __restrict__ x, float* __restrict__ out) {
    const int dif = blockIdx.x;             // 0..255  output row within image
    const int bc  = blockIdx.y;             // 0..511  fused (B,C) image index
    const int tid = threadIdx.x;            // 0..255

    const float* __restrict__ src = x + (size_t)bc * (H * W);
    float* __restrict__ dst = out + ((size_t)bc * NROWS + dif) * NCOLS;

    const int i  = 255 - dif;
    const int el = 2 * dif + 1;

    // Warm the two fully-consumed rows of this ring (rows i and j = 511-i):
    // 16 cachelines (128B) each -> threads 0..31, emits global_prefetch_b8.
    if (tid < 16) {
        __builtin_prefetch(src + i * W + tid * 32, 0, 3);
    } else if (tid < 32) {
        __builtin_prefetch(src + (511 - i) * W + (tid - 16) * 32, 0, 3);
    }

    // Each thread produces 8 output cols as two float4 groups; within a wave
    // the 32 lanes write one contiguous 512B burst (global_store_b128 each).
#pragma unroll
    for (int half = 0; half < 2; ++half) {
        const int cbase = half * 1024 + tid * 4;
        float v[4];
#pragma unroll
        for (int k = 0; k < 4; ++k) {
            const int c    = cbase + k;
            const int ceff = (c >= RINGLEN) ? (c - RINGLEN) : c;  // tail wrap
            v[k] = src[ring_src(i, el, ceff)];
        }
        // Output is write-once streaming: non-temporal store (TH=NT) keeps the
        // 192MB L2 free to hold the reused source image.
        v4f pack = {v[0], v[1], v[2], v[3]};
        __builtin_nontemporal_store(pack, reinterpret_cast<v4f*>(dst + cbase));
    }
}

extern "C" void kernel_launch(void* const* d_in, const int* in_sizes, int n_in,
                              void* d_out, int out_size, void* d_ws, size_t ws_size,
                              hipStream_t stream) {
    (void)n_in; (void)out_size; (void)d_ws; (void)ws_size;
    const float* x  = (const float*)d_in[0];
    float* out      = (float*)d_out;

    // B*C derived from the input's flat size (reference: 16*32 = 512 images).
    const int bcTotal = in_sizes[0] / (H * W);

    dim3 grid(NROWS, bcTotal);
    dim3 block(256);
    SqRL_ring_gather<<<grid, block, 0, stream>>>(x, out);
}